// LNN_19713899889340
// MI455X (gfx1250) — compile-verified
//
#include <hip/hip_runtime.h>

#define D_ 8
#define HID 200
#define NB 16        // samples per workgroup (one WMMA M-tile)
#define LDA 212      // stream-buffer row stride in floats (conflict-free)
#define NT 13        // ceil(200/16) N tiles
#define EPS_ 0.1f
#define BLK 128      // 4 waves
#define KP 20        // K panel size (10 panels x 20 = 200)
#define NPAN (HID / KP)
#define LDP 22       // panel row stride in floats (conflict-free, b64-aligned)
#define PROWS 208    // padded N rows in panel
#define PSZ (PROWS * LDP)
#define PELEM (PROWS * KP)   // 4160
#define PREG 33      // ceil(PELEM / BLK)

typedef __attribute__((ext_vector_type(2))) float v2f;
typedef __attribute__((ext_vector_type(8))) float v8f;

#if __has_builtin(__builtin_amdgcn_global_load_async_to_lds_b32)
#define ASYNC_LDS 1
typedef __attribute__((address_space(1))) int* gas1i_;   // global src (prints as __device__ int*)
typedef __attribute__((address_space(3))) int* las3i_;   // LDS dst
#endif

__device__ __forceinline__ void wait_async0_() {
#if defined(ASYNC_LDS)
#if __has_builtin(__builtin_amdgcn_s_wait_asynccnt)
  __builtin_amdgcn_s_wait_asynccnt(0);
#else
  asm volatile("s_wait_asynccnt 0x0" ::: "memory");
#endif
#endif
}

__device__ __forceinline__ float sigmoidf_(float x) { return 1.0f / (1.0f + expf(-x)); }
__device__ __forceinline__ float softplusf_(float x) {
  return fmaxf(x, 0.0f) + log1pf(expf(-fabsf(x)));
}

// ---- W K-panel staging ----
// Panel holds B(k,n), k in [k0,k0+KP), n in [0,208) (rows n>=HID pre-zeroed),
// stored Bp[n*LDP + kk] so fragments read b64 pairs (kk, kk+1).

#if defined(ASYNC_LDS)
// CDNA5 async DMA: global -> LDS, no VGPR round trip, tracked by ASYNCcnt.
template <bool TRANS>
__device__ __forceinline__ void panel_stage_async(const float* __restrict__ W, int k0,
                                                  float* __restrict__ Bp) {
  const int tid = threadIdx.x;
#pragma unroll
  for (int i = 0; i < PREG; ++i) {
    int idx = tid + i * BLK;
    if (idx < PELEM) {
      int n, kk;
      if (TRANS) { n = idx / KP; kk = idx - n * KP; }       // kk fastest: coalesced runs
      else       { kk = idx / PROWS; n = idx - kk * PROWS; } // n fastest: coalesced rows
      if (n < HID) {
        const float* gp = TRANS ? (W + n * HID + (k0 + kk))
                                : (W + (k0 + kk) * HID + n);
        float* lp = Bp + n * LDP + kk;
        __builtin_amdgcn_global_load_async_to_lds_b32((gas1i_)gp, (las3i_)lp, 0, 0);
      }
    }
  }
}
#else
template <bool TRANS>
__device__ __forceinline__ void panel_prefetch(const float* __restrict__ W, int k0,
                                               float* __restrict__ r) {
  const int tid = threadIdx.x;
#pragma unroll
  for (int i = 0; i < PREG; ++i) {
    int idx = tid + i * BLK;
    float v = 0.0f;
    if (idx < PELEM) {
      if (TRANS) {
        int n = idx / KP, kk = idx - n * KP;
        if (n < HID) v = W[n * HID + (k0 + kk)];
      } else {
        int kk = idx / PROWS, n = idx - kk * PROWS;
        if (n < HID) v = W[(k0 + kk) * HID + n];
      }
    }
    r[i] = v;
  }
}
template <bool TRANS>
__device__ __forceinline__ void panel_commit(float* __restrict__ Bp,
                                             const float* __restrict__ r) {
  const int tid = threadIdx.x;
#pragma unroll
  for (int i = 0; i < PREG; ++i) {
    int idx = tid + i * BLK;
    if (idx < PELEM) {
      int n, kk;
      if (TRANS) { n = idx / KP; kk = idx - n * KP; }
      else       { kk = idx / PROWS; n = idx - kk * PROWS; }
      Bp[n * LDP + kk] = r[i];
    }
  }
}
#endif

// Out[q][m][n] = sum_k A[q][m][k] * B(k,n). M=16 samples, K=200, N tiles = NT.
// Wave w owns tiles nt = w, w+4, w+8 (+12 for w==0); accumulators live across the
// whole K loop. Per K-step: NQ A-frag b64 loads (shared over tiles) + up to 4 B-frag
// b64 loads (shared over streams) feed 4*NQ independent WMMAs.
template <int NQ, bool TRANS>
__device__ __forceinline__ void gemm_tiles(const float* __restrict__ Abase,
                                           float* __restrict__ Obase,
                                           const float* __restrict__ W,
                                           float* __restrict__ Bpan) {  // [2][PSZ]
  const int tid = threadIdx.x;
  const int lane = tid & 31;
  const int wave = tid >> 5;
  const int ln = lane & 15;
  const int hi = lane >> 4;
  const int kb = hi * 2;
  const int ntl = (wave == 0) ? 4 : 3;  // tiles this wave owns (NT = 13)

  v8f acc[4][NQ];
#pragma unroll
  for (int t = 0; t < 4; ++t)
#pragma unroll
    for (int q = 0; q < NQ; ++q) {
      v8f zz = {0.f, 0.f, 0.f, 0.f, 0.f, 0.f, 0.f, 0.f};
      acc[t][q] = zz;
    }

#if defined(ASYNC_LDS)
  panel_stage_async<TRANS>(W, 0, Bpan);
  wait_async0_();
  __syncthreads();
#else
  float r[PREG];
  panel_prefetch<TRANS>(W, 0, r);
  panel_commit<TRANS>(Bpan, r);
  __syncthreads();
#endif

  for (int p = 0; p < NPAN; ++p) {
    const float* bufc = Bpan + (p & 1) * PSZ;
#if defined(ASYNC_LDS)
    if (p < NPAN - 1) panel_stage_async<TRANS>(W, (p + 1) * KP, Bpan + ((p + 1) & 1) * PSZ);
#else
    if (p < NPAN - 1) panel_prefetch<TRANS>(W, (p + 1) * KP, r);
#endif

#pragma unroll
    for (int ks = 0; ks < KP / 4; ++ks) {
      const int k0 = p * KP + ks * 4;
      v2f a[NQ];
#pragma unroll
      for (int q = 0; q < NQ; ++q) {
        const float* A = Abase + q * (NB * LDA) + ln * LDA + k0 + kb;
        a[q].x = A[0]; a[q].y = A[1];
      }
#pragma unroll
      for (int t = 0; t < 4; ++t) {
        if (t < ntl) {
          const int n0 = (wave + 4 * t) * 16;
          const float* Bf = bufc + (n0 + ln) * LDP + ks * 4 + kb;
          v2f b; b.x = Bf[0]; b.y = Bf[1];
#pragma unroll
          for (int q = 0; q < NQ; ++q)
            acc[t][q] = __builtin_amdgcn_wmma_f32_16x16x4_f32(
                false, a[q], false, b, (short)0, acc[t][q], false, false);
        }
      }
    }
#if defined(ASYNC_LDS)
    if (p < NPAN - 1) wait_async0_();
#else
    if (p < NPAN - 1) panel_commit<TRANS>(Bpan + ((p + 1) & 1) * PSZ, r);
#endif
    __syncthreads();
  }

#pragma unroll
  for (int t = 0; t < 4; ++t) {
    if (t < ntl) {
      const int n0 = (wave + 4 * t) * 16;
#pragma unroll
      for (int q = 0; q < NQ; ++q) {
        float* O = Obase + q * (NB * LDA);
#pragma unroll
        for (int r8 = 0; r8 < 8; ++r8) O[(r8 + hi * 8) * LDA + n0 + ln] = acc[t][q][r8];
      }
    }
  }
}

// Out[m][n] (n<16) = sum_k A[m][k] * W0lds[n][k] == A @ W0^T (single tile, one wave).
__device__ __forceinline__ void gemm_hcol(const float* __restrict__ A,
                                          const float* __restrict__ W0l,
                                          float* __restrict__ Out) {
  const int lane = threadIdx.x & 31;
  const int ln = lane & 15;
  const int hi = lane >> 4;
  const int kb = hi * 2;
  v8f acc = {0.f, 0.f, 0.f, 0.f, 0.f, 0.f, 0.f, 0.f};
  for (int k0 = 0; k0 < HID; k0 += 4) {
    v2f b, a;
    b.x = W0l[ln * LDA + k0 + kb];
    b.y = W0l[ln * LDA + k0 + kb + 1];
    a.x = A[ln * LDA + k0 + kb];
    a.y = A[ln * LDA + k0 + kb + 1];
    acc = __builtin_amdgcn_wmma_f32_16x16x4_f32(false, a, false, b,
                                                (short)0, acc, false, false);
  }
#pragma unroll
  for (int r = 0; r < 8; ++r) Out[(r + hi * 8) * 16 + ln] = acc[r];
}

__global__ __launch_bounds__(BLK) void lnn_dyn_kernel(
    const float* __restrict__ z,
    const float* __restrict__ W0, const float* __restrict__ b0,
    const float* __restrict__ W1, const float* __restrict__ b1,
    const float* __restrict__ W2, const float* __restrict__ b2,
    const float* __restrict__ W3,
    float* __restrict__ out, int N) {
  __shared__ float sBufA[4][NB][LDA];
  __shared__ float sBufB[4][NB][LDA];
  __shared__ float sBufC[4][NB][LDA];
  __shared__ float sS1[NB][LDA], sS2[NB][LDA], sS3[NB][LDA];  // sigmoid(u_l)
  __shared__ float sG1[NB][LDA], sG2[NB][LDA];                // primal cotangents on h1,h2
  __shared__ float sW0[16][LDA];                               // W0 (16 x 200)
  __shared__ float sW3[224];                                   // W3 column
  __shared__ float sHcol[8][NB][16];                           // H[:, 8+j] per sample
  __shared__ float sGz[NB][16];                                // primal grad wrt z
  __shared__ float sZ[NB][16];
  __shared__ float sBpan[2][PSZ];                              // W K-panel double buffer

  const int tid = threadIdx.x;
  const int blk = blockIdx.x;

  // ---- stage inputs; pre-zero panel pad rows (n in [HID,PROWS)) once ----
  for (int i = tid; i < NB * 16; i += BLK) {
    int m = i >> 4, c = i & 15;
    int sg = blk * NB + m; if (sg >= N) sg = N - 1;
    sZ[m][c] = z[sg * 16 + c];
  }
  for (int i = tid; i < 16 * HID; i += BLK) {
    int r = i / HID, c = i - r * HID;
    sW0[r][c] = W0[i];
  }
  for (int i = tid; i < 224; i += BLK) sW3[i] = (i < HID) ? W3[i] : 0.0f;
  for (int i = tid; i < 2 * (PROWS - HID) * LDP; i += BLK) {
    int b = i / ((PROWS - HID) * LDP), rrem = i - b * ((PROWS - HID) * LDP);
    sBpan[b][HID * LDP + rrem] = 0.0f;
  }
  __syncthreads();

  // ================= PRIMAL PASS =================
  // P1: u1 = z @ W0 + b0 (tiny, VALU); keep s1, h1
  for (int idx = tid; idx < NB * HID; idx += BLK) {
    int m = idx / HID, c = idx - m * HID;
    float u = b0[c];
#pragma unroll
    for (int i = 0; i < 16; ++i) u += sZ[m][i] * sW0[i][c];
    sS1[m][c] = sigmoidf_(u);
    sBufA[0][m][c] = softplusf_(u);  // h1
  }
  __syncthreads();
  // P2: u2 = h1 @ W1
  gemm_tiles<1, false>(&sBufA[0][0][0], &sBufB[0][0][0], W1, &sBpan[0][0]);
  __syncthreads();
  for (int idx = tid; idx < NB * HID; idx += BLK) {
    int m = idx / HID, c = idx - m * HID;
    float u = sBufB[0][m][c] + b1[c];
    sS2[m][c] = sigmoidf_(u);
    sBufB[0][m][c] = softplusf_(u);  // h2
  }
  __syncthreads();
  // P3: u3 = h2 @ W2 ; c3 = w3 * s3
  gemm_tiles<1, false>(&sBufB[0][0][0], &sBufA[0][0][0], W2, &sBpan[0][0]);
  __syncthreads();
  for (int idx = tid; idx < NB * HID; idx += BLK) {
    int m = idx / HID, c = idx - m * HID;
    float u = sBufA[0][m][c] + b2[c];
    float s3 = sigmoidf_(u);
    sS3[m][c] = s3;
    sBufA[0][m][c] = sW3[c] * s3;  // c3
  }
  __syncthreads();
  // P4: g2 = c3 @ W2^T ; c2 = g2 * s2
  gemm_tiles<1, true>(&sBufA[0][0][0], &sBufC[0][0][0], W2, &sBpan[0][0]);
  __syncthreads();
  for (int idx = tid; idx < NB * HID; idx += BLK) {
    int m = idx / HID, c = idx - m * HID;
    float g2 = sBufC[0][m][c];
    sG2[m][c] = g2;
    sBufA[0][m][c] = g2 * sS2[m][c];  // c2
  }
  __syncthreads();
  // P5: g1 = c2 @ W1^T ; c1 = g1 * s1
  gemm_tiles<1, true>(&sBufA[0][0][0], &sBufB[0][0][0], W1, &sBpan[0][0]);
  __syncthreads();
  for (int idx = tid; idx < NB * HID; idx += BLK) {
    int m = idx / HID, c = idx - m * HID;
    float g1 = sBufB[0][m][c];
    sG1[m][c] = g1;
    sBufA[0][m][c] = g1 * sS1[m][c];  // c1
  }
  __syncthreads();
  // P6: gz = c1 @ W0^T (wave 0)
  if (tid < 32) gemm_hcol(&sBufA[0][0][0], &sW0[0][0], &sGz[0][0]);
  __syncthreads();

  // ================= TANGENT PASSES (2 groups x 4 streams) =================
  for (int g = 0; g < 2; ++g) {
    // T1: tdot1_j = s1 * W0[8+jt,:]
    for (int q = 0; q < 4; ++q) {
      int jt = 4 * g + q;
      for (int idx = tid; idx < NB * HID; idx += BLK) {
        int m = idx / HID, c = idx - m * HID;
        sBufA[q][m][c] = sS1[m][c] * sW0[8 + jt][c];
      }
    }
    __syncthreads();
    // T2: udot2 = tdot1 @ W1 ; tdot2 = s2 * udot2 (in place)
    gemm_tiles<4, false>(&sBufA[0][0][0], &sBufB[0][0][0], W1, &sBpan[0][0]);
    __syncthreads();
    for (int q = 0; q < 4; ++q)
      for (int idx = tid; idx < NB * HID; idx += BLK) {
        int m = idx / HID, c = idx - m * HID;
        sBufB[q][m][c] *= sS2[m][c];
      }
    __syncthreads();
    // T3: udot3 = tdot2 @ W2 ; cdot3 = w3 * s3*(1-s3) * udot3 (in place)
    gemm_tiles<4, false>(&sBufB[0][0][0], &sBufA[0][0][0], W2, &sBpan[0][0]);
    __syncthreads();
    for (int q = 0; q < 4; ++q)
      for (int idx = tid; idx < NB * HID; idx += BLK) {
        int m = idx / HID, c = idx - m * HID;
        float s3 = sS3[m][c];
        sBufA[q][m][c] = sW3[c] * s3 * (1.0f - s3) * sBufA[q][m][c];
      }
    __syncthreads();
    // T4: gdot2 = cdot3 @ W2^T ; cdot2 = gdot2*s2 + g2*(1-s2)*tdot2
    gemm_tiles<4, true>(&sBufA[0][0][0], &sBufC[0][0][0], W2, &sBpan[0][0]);
    __syncthreads();
    for (int q = 0; q < 4; ++q)
      for (int idx = tid; idx < NB * HID; idx += BLK) {
        int m = idx / HID, c = idx - m * HID;
        float s2 = sS2[m][c];
        sBufC[q][m][c] = sBufC[q][m][c] * s2 + sG2[m][c] * (1.0f - s2) * sBufB[q][m][c];
      }
    __syncthreads();
    // T5: gdot1 = cdot2 @ W1^T ; cdot1 = gdot1*s1 + g1*s1*(1-s1)*W0[8+jt,:]
    gemm_tiles<4, true>(&sBufC[0][0][0], &sBufA[0][0][0], W1, &sBpan[0][0]);
    __syncthreads();
    for (int q = 0; q < 4; ++q) {
      int jt = 4 * g + q;
      for (int idx = tid; idx < NB * HID; idx += BLK) {
        int m = idx / HID, c = idx - m * HID;
        float s1 = sS1[m][c];
        sBufA[q][m][c] = sBufA[q][m][c] * s1 +
                         sG1[m][c] * s1 * (1.0f - s1) * sW0[8 + jt][c];
      }
    }
    __syncthreads();
    // T6: Hcol_jt = cdot1 @ W0^T (wave q handles stream q)
    {
      int w = tid >> 5;
      gemm_hcol(&sBufA[w][0][0], &sW0[0][0], &sHcol[4 * g + w][0][0]);
    }
    __syncthreads();
  }

  // ================= SOLVE (one sample per thread) =================
  if (tid < NB) {
    const int m = tid;
    const int sg = blk * NB + m;
    float vv[8], Mm[8][8], Fv[8], a8[8];
#pragma unroll
    for (int k = 0; k < 8; ++k) vv[k] = sZ[m][8 + k];
    for (int r = 0; r < 8; ++r) {
      for (int c = 0; c < 8; ++c) Mm[r][c] = sHcol[c][m][8 + r];  // H[8+r][8+c]
      Mm[r][r] += 2.0f * EPS_;
    }
    for (int i = 0; i < 8; ++i) {
      float acc = sGz[m][i];                                      // g[:8]
      for (int k = 0; k < 8; ++k) acc -= sHcol[i][m][k] * vv[k];  // C[i][k]*v[k]
      Fv[i] = acc;
    }
    // Gaussian elimination with partial pivoting
    for (int col = 0; col < 8; ++col) {
      int p = col; float best = fabsf(Mm[col][col]);
      for (int r = col + 1; r < 8; ++r) {
        float t = fabsf(Mm[r][col]);
        if (t > best) { best = t; p = r; }
      }
      if (p != col) {
        for (int c = col; c < 8; ++c) { float t = Mm[col][c]; Mm[col][c] = Mm[p][c]; Mm[p][c] = t; }
        float t = Fv[col]; Fv[col] = Fv[p]; Fv[p] = t;
      }
      float inv = 1.0f / Mm[col][col];
      for (int r = col + 1; r < 8; ++r) {
        float f = Mm[r][col] * inv;
        for (int c = col; c < 8; ++c) Mm[r][c] -= f * Mm[col][c];
        Fv[r] -= f * Fv[col];
      }
    }
    for (int r = 7; r >= 0; --r) {
      float s = Fv[r];
      for (int c = r + 1; c < 8; ++c) s -= Mm[r][c] * a8[c];
      a8[r] = s / Mm[r][r];
    }
    if (sg < N) {
      float* orow = out + sg * 16;
#pragma unroll
      for (int k = 0; k < 8; ++k) { orow[k] = vv[k]; orow[8 + k] = a8[k]; }
    }
  }
}

extern "C" void kernel_launch(void* const* d_in, const int* in_sizes, int n_in,
                              void* d_out, int out_size, void* d_ws, size_t ws_size,
                              hipStream_t stream) {
  // inputs: 0=t, 1=z, 2=W0, 3=b0, 4=W1, 5=b1, 6=W2, 7=b2, 8=W3, 9=b3
  const float* z  = (const float*)d_in[1];
  const float* W0 = (const float*)d_in[2];
  const float* b0 = (const float*)d_in[3];
  const float* W1 = (const float*)d_in[4];
  const float* b1 = (const float*)d_in[5];
  const float* W2 = (const float*)d_in[6];
  const float* b2 = (const float*)d_in[7];
  const float* W3 = (const float*)d_in[8];
  float* out = (float*)d_out;
  const int N = in_sizes[1] / 16;          // 2*D per sample
  const int grid = (N + NB - 1) / NB;
  lnn_dyn_kernel<<<grid, BLK, 0, stream>>>(z, W0, b0, W1, b1, W2, b2, W3, out, N);
}